// DeStripeModel_34196529610979
// MI455X (gfx1250) — compile-verified
//
#include <hip/hip_runtime.h>
#include <math.h>

// ---------------- problem constants ----------------
#define MDIM 769
#define NDIM 769
#define MN   591361            // M*N
#define KC   295680            // K = (M*N)/2
#define INC  16
#define NNEI 16
#define NNI  147840            // K/2
#define KS   29
#define PADD 800               // 769 padded to 16*50 = 32*25 for 2-wide WMMA tiling
#define PP   (PADD*PADD)

typedef __attribute__((ext_vector_type(2))) float v2f;
typedef __attribute__((ext_vector_type(8))) float v8f;

// ---------------- complex helpers ----------------
struct c32 { float x, y; };
__device__ __forceinline__ c32 cmul(c32 a, c32 b){
  return { fmaf(a.x,b.x,-a.y*b.y), fmaf(a.x,b.y, a.y*b.x) };
}
__device__ __forceinline__ c32 cadd(c32 a, c32 b){ return {a.x+b.x, a.y+b.y}; }
__device__ __forceinline__ c32 cmadd(c32 acc, c32 a, c32 b){
  acc.x = fmaf(a.x, b.x, fmaf(-a.y, b.y, acc.x));
  acc.y = fmaf(a.x, b.y, fmaf( a.y, b.x, acc.y));
  return acc;
}
__device__ __forceinline__ c32 crelu(c32 a){ return { fmaxf(a.x,0.f), fmaxf(a.y,0.f) }; }
__device__ __forceinline__ c32 cscale(c32 a, float s){ return {a.x*s, a.y*s}; }

// 16x16 complex linear out = W*in + b (W row-major [out][in]), optional crelu
__device__ __forceinline__ void clin16(const c32* __restrict__ W, const c32* __restrict__ b,
                                       const c32* in, c32* out, bool relu){
  #pragma unroll
  for (int f=0; f<16; ++f){
    c32 acc = b[f];
    #pragma unroll
    for (int g=0; g<16; ++g) acc = cmadd(acc, W[f*16+g], in[g]);
    out[f] = relu ? crelu(acc) : acc;
  }
}

// ================= kernel 1: p = ResLearning(1,16) =================
__global__ void __launch_bounds__(256)
k_pointwise(const c32* __restrict__ Xf,
            const c32* __restrict__ f1W, const c32* __restrict__ f1b,
            const c32* __restrict__ f2W, const c32* __restrict__ f2b,
            const c32* __restrict__ skW, const c32* __restrict__ skb,
            c32* __restrict__ Xfp){
  __shared__ c32 sW[256];
  __shared__ c32 s1W[16], s1b[16], s2b[16], sSW[16], sSb[16];
  int t = threadIdx.x;
  sW[t] = f2W[t];
  if (t < 16){ s1W[t]=f1W[t]; s1b[t]=f1b[t]; s2b[t]=f2b[t]; sSW[t]=skW[t]; sSb[t]=skb[t]; }
  __syncthreads();
  int i = blockIdx.x*256 + t;
  if (i >= KC) return;
  c32 xin = Xf[i];
  c32 h[16];
  #pragma unroll
  for (int f=0; f<16; ++f) h[f] = crelu(cmadd(s1b[f], s1W[f], xin));   // crelu(f1)
  #pragma unroll
  for (int f=0; f<16; ++f){
    c32 acc = cadd(cmadd(s2b[f], sSW[f], xin), sSb[f]);                // f2.b + skip
    #pragma unroll
    for (int g=0; g<16; ++g) acc = cmadd(acc, sW[f*16+g], h[g]);
    Xfp[(size_t)i*16 + f] = crelu(acc);
  }
}

// ================= kernel 2: gather + edge/latent/mlp chain -> z =================
__global__ void __launch_bounds__(256)
k_main(const c32* __restrict__ Xfp, c32* __restrict__ zout,
       const int* __restrict__ NI, const float* __restrict__ wcoef,
       const int* __restrict__ hmask, const int* __restrict__ hind,
       const c32* __restrict__ e1W, const c32* __restrict__ e1b,
       const c32* __restrict__ e2W, const c32* __restrict__ e2b,
       const c32* __restrict__ l1W, const c32* __restrict__ l1b,
       const c32* __restrict__ l2W, const c32* __restrict__ l2b,
       const c32* __restrict__ m1W, const c32* __restrict__ m1b,
       const c32* __restrict__ m2W, const c32* __restrict__ m2b,
       const c32* __restrict__ m3W, const c32* __restrict__ m3b){
  __shared__ c32 sW[6][256];
  __shared__ c32 sB[6][16];
  __shared__ c32 sM3W[16];
  __shared__ c32 sM3b;
  int t = threadIdx.x;
  sW[0][t]=e1W[t]; sW[1][t]=e2W[t]; sW[2][t]=l1W[t];
  sW[3][t]=l2W[t]; sW[4][t]=m1W[t]; sW[5][t]=m2W[t];
  if (t < 16){
    sB[0][t]=e1b[t]; sB[1][t]=e2b[t]; sB[2][t]=l1b[t];
    sB[3][t]=l2b[t]; sB[4][t]=m1b[t]; sB[5][t]=m2b[t];
    sM3W[t]=m3W[t];
  }
  if (t == 0) sM3b = m3b[0];
  __syncthreads();

  int i = blockIdx.x*256 + t;
  if (i >= KC) return;

  // analytic TVX/TVY/EIG at fftshifted flat index i
  int u = i / NDIM, v = i - u*NDIM;
  int u0 = (u + 385) % 769, v0 = (v + 385) % 769;    // (idx - 384) mod 769
  float thv = 6.28318530717958647692f * (float)v0 * (1.0f/769.0f);
  float thu = 6.28318530717958647692f * (float)u0 * (1.0f/769.0f);
  float sv, cv, su, cu;
  __sincosf(thv, &sv, &cv);
  __sincosf(thu, &su, &cu);
  c32 tvx = {1.f - cv, sv}, tvy = {1.f - cu, su};
  c32 itvx = {tvx.x, -tvx.y}, itvy = {tvy.x, -tvy.y};
  float eig = 1.f / ((2.f - 2.f*cv) + (2.f - 2.f*cu));

  // xtv = concat(Xfcx, Xfp[hier_mask])[hier_ind[i]]
  c32 X[16];
  int hi = hind[i];
  if (hi < NNI){
    #pragma unroll
    for (int f=0; f<16; ++f) X[f] = {0.f, 0.f};
    for (int j=0; j<NNEI; ++j){
      int idx  = NI[j*NNI + hi];
      float wj = wcoef[j*NNI + hi];
      const c32* row = &Xfp[(size_t)idx*16];
      #pragma unroll
      for (int f=0; f<16; ++f){
        X[f].x = fmaf(wj, row[f].x, X[f].x);
        X[f].y = fmaf(wj, row[f].y, X[f].y);
      }
    }
  } else {
    int idx = hmask[hi - NNI];
    const c32* row = &Xfp[(size_t)idx*16];
    #pragma unroll
    for (int f=0; f<16; ++f) X[f] = row[f];
  }

  c32 Y[16];
  { const c32* row = &Xfp[(size_t)i*16];
    #pragma unroll
    for (int f=0; f<16; ++f) Y[f] = cmul(row[f], tvy); }   // y = Xfp*TVY
  #pragma unroll
  for (int f=0; f<16; ++f) X[f] = cmul(X[f], tvx);         // x = xtv*TVX

  c32 T[16], EX[16], EY[16];
  clin16(sW[0], sB[0], X, T, true);                        // e1 + crelu
  clin16(sW[1], sB[1], T, EX, false);                      // e2
  #pragma unroll
  for (int f=0; f<16; ++f) EX[f] = crelu(cmul(EX[f], itvx));
  clin16(sW[0], sB[0], Y, T, true);
  clin16(sW[1], sB[1], T, EY, false);
  #pragma unroll
  for (int f=0; f<16; ++f) EX[f] = cadd(EX[f], crelu(cmul(EY[f], itvy)));
  clin16(sW[2], sB[2], EX, T, true);                       // l1 + crelu
  clin16(sW[3], sB[3], T, EX, false);                      // l2
  #pragma unroll
  for (int f=0; f<16; ++f) EX[f] = crelu(cscale(EX[f], eig));
  clin16(sW[4], sB[4], EX, T, true);                       // m1 + crelu
  clin16(sW[5], sB[5], T, EX, true);                       // m2 + crelu
  c32 acc = sM3b;
  #pragma unroll
  for (int g=0; g<16; ++g) acc = cmadd(acc, sM3W[g], EX[g]);
  zout[i] = acc;
}

// ================= deterministic Xd sum (two stage) =================
__global__ void __launch_bounds__(256)
k_sum_stage1(const float* __restrict__ Xd, float* __restrict__ part){
  __shared__ float sm[256];
  const int chunk = (MN + 255) / 256;              // 2311
  int b = blockIdx.x;
  int lo = b * chunk;
  int hiX = lo + chunk; if (hiX > MN) hiX = MN;
  float acc = 0.f;
  for (int i = lo + threadIdx.x; i < hiX; i += 256) acc += Xd[i];
  sm[threadIdx.x] = acc;
  __syncthreads();
  for (int s = 128; s > 0; s >>= 1){
    if (threadIdx.x < s) sm[threadIdx.x] += sm[threadIdx.x + s];
    __syncthreads();
  }
  if (threadIdx.x == 0) part[b] = sm[0];
}
__global__ void __launch_bounds__(256)
k_sum_stage2(const float* __restrict__ part, float* __restrict__ S){
  __shared__ float sm[256];
  sm[threadIdx.x] = part[threadIdx.x];
  __syncthreads();
  for (int s = 128; s > 0; s >>= 1){
    if (threadIdx.x < s) sm[threadIdx.x] += sm[threadIdx.x + s];
    __syncthreads();
  }
  if (threadIdx.x == 0) S[0] = sm[0];
}

// ================= IDFT matrix E[m,u] = exp(+2*pi*i*m*u/769), zero-padded =================
__global__ void __launch_bounds__(256)
k_dft_e(float* __restrict__ Er, float* __restrict__ Ei){
  int idx = blockIdx.x*256 + threadIdx.x;
  if (idx >= PP) return;
  int m = idx / PADD, u = idx - m*PADD;
  float er = 0.f, ei = 0.f;
  if (m < MDIM && u < NDIM){
    int r = (m * u) % 769;
    float th = 6.28318530717958647692f * (float)r * (1.0f/769.0f);
    __sincosf(th, &ei, &er);
  }
  Er[idx] = er; Ei[idx] = ei;
}

// ================= assemble ifftshift(zf) into padded planes =================
__global__ void __launch_bounds__(256)
k_build_zf(const c32* __restrict__ z, const float* __restrict__ S,
           const float* __restrict__ b1W, const float* __restrict__ b1b,
           const float* __restrict__ b2W, const float* __restrict__ b2b,
           const float* __restrict__ b3W, const float* __restrict__ b3b,
           float* __restrict__ Zr, float* __restrict__ Zi){
  int idx = blockIdx.x*256 + threadIdx.x;
  if (idx >= PP) return;
  int up = idx / PADD, vp = idx - up*PADD;
  float re = 0.f, im = 0.f;
  if (up < MDIM && vp < NDIM){
    int s = ((up + 384) % 769) * 769 + ((vp + 384) % 769);  // ifftshift source
    if (s < KC){ re = z[s].x; im = z[s].y; }
    else if (s == KC){
      // dc = baseMLP(1) * sum(Xd)
      float h1[16];
      #pragma unroll
      for (int f=0; f<16; ++f) h1[f] = fmaxf(b1W[f] + b1b[f], 0.f);
      float base = b3b[0];
      #pragma unroll
      for (int f=0; f<16; ++f){
        float a2 = b2b[f];
        #pragma unroll
        for (int g=0; g<16; ++g) a2 = fmaf(b2W[f*16+g], h1[g], a2);
        base = fmaf(b3W[f], fmaxf(a2, 0.f), base);
      }
      re = base * S[0]; im = 0.f;
    } else {
      int s2 = MN - 1 - s;               // conj(z[::-1])
      re = z[s2].x; im = -z[s2].y;
    }
  }
  Zr[idx] = re; Zi[idx] = im;
}

// ============================================================================
// Complex GEMM stages for O = E * Z * E = E * (Z * E), all fragments loaded as
// contiguous b64 pairs:
//   stage 1:  U = Z * E, B-side uses E symmetry  (B[k][n] = E[n][k], row read),
//             result stored TRANSPOSED (Ut) with contiguous float4 stores.
//   stage 2:  O = E * U, B-side reads Ut rows    (B[k][n] = Ut[n][k], row read).
// One wave computes a 16x32 tile (2 column tiles share the A fragments):
// 8 wmma per k-step for 6 b64 loads.
// Fragment layouts per ISA 7.12.2:
//   A 16x4 : lanes 0-15 row=lane  K={k0,k0+1}; lanes 16-31 row=lane-16 K={k0+2,k0+3}
//   B 4x16 : lanes 0-15 K={k0,k0+1} col=lane ; lanes 16-31 K={k0+2,k0+3} col=lane-16
//   C/D    : VGPR v = row v (lanes 0-15) / row v+8 (lanes 16-31), col = lane&15
// ============================================================================
__global__ void __launch_bounds__(32)
k_cgemm_s1(const float* __restrict__ Zr, const float* __restrict__ Zi,
           const float* __restrict__ Er, const float* __restrict__ Ei,
           float* __restrict__ Utr, float* __restrict__ Uti){
  const int lane = threadIdx.x;
  const int rm = blockIdx.y * 16;
  const int cn = blockIdx.x * 32;
  const int half = lane >> 4;
  const int l15  = lane & 15;
  const size_t arow  = rm + l15;
  const size_t bcol0 = cn + l15;
  const size_t bcol1 = cn + 16 + l15;
  v8f z8 = {0.f,0.f,0.f,0.f,0.f,0.f,0.f,0.f};
  v8f pp0=z8, qq0=z8, rr0=z8, ss0=z8;       // col tile 0: ArBr, AiBi, ArBi, AiBr
  v8f pp1=z8, qq1=z8, rr1=z8, ss1=z8;       // col tile 1
  const float* ArRow = Zr + arow*PADD;
  const float* AiRow = Zi + arow*PADD;
  const float* Br0   = Er + bcol0*PADD;     // E symmetric: row bcol == col bcol
  const float* Bi0   = Ei + bcol0*PADD;
  const float* Br1   = Er + bcol1*PADD;
  const float* Bi1   = Ei + bcol1*PADD;
  for (int k0 = 0; k0 < PADD; k0 += 4){
    const int ka = k0 + half*2;
    __builtin_prefetch(ArRow + ka + 64, 0, 1);
    v2f ar  = { ArRow[ka], ArRow[ka+1] };
    v2f ai  = { AiRow[ka], AiRow[ka+1] };
    v2f br0 = { Br0[ka],   Br0[ka+1]   };
    v2f bi0 = { Bi0[ka],   Bi0[ka+1]   };
    v2f br1 = { Br1[ka],   Br1[ka+1]   };
    v2f bi1 = { Bi1[ka],   Bi1[ka+1]   };
    pp0 = __builtin_amdgcn_wmma_f32_16x16x4_f32(false, ar, false, br0, (short)0, pp0, false, false);
    qq0 = __builtin_amdgcn_wmma_f32_16x16x4_f32(false, ai, false, bi0, (short)0, qq0, false, false);
    rr0 = __builtin_amdgcn_wmma_f32_16x16x4_f32(false, ar, false, bi0, (short)0, rr0, false, false);
    ss0 = __builtin_amdgcn_wmma_f32_16x16x4_f32(false, ai, false, br0, (short)0, ss0, false, false);
    pp1 = __builtin_amdgcn_wmma_f32_16x16x4_f32(false, ar, false, br1, (short)0, pp1, false, false);
    qq1 = __builtin_amdgcn_wmma_f32_16x16x4_f32(false, ai, false, bi1, (short)0, qq1, false, false);
    rr1 = __builtin_amdgcn_wmma_f32_16x16x4_f32(false, ar, false, bi1, (short)0, rr1, false, false);
    ss1 = __builtin_amdgcn_wmma_f32_16x16x4_f32(false, ai, false, br1, (short)0, ss1, false, false);
  }
  // transposed store: lane holds rows rbase..rbase+7 of column bcol -> contiguous
  const int rbase = rm + half*8;
  float re[8], im[8];
  #pragma unroll
  for (int vg=0; vg<8; ++vg){ re[vg] = pp0[vg]-qq0[vg]; im[vg] = rr0[vg]+ss0[vg]; }
  {
    float* dr = Utr + bcol0*PADD + rbase;
    float* di = Uti + bcol0*PADD + rbase;
    ((float4*)dr)[0] = make_float4(re[0],re[1],re[2],re[3]);
    ((float4*)dr)[1] = make_float4(re[4],re[5],re[6],re[7]);
    ((float4*)di)[0] = make_float4(im[0],im[1],im[2],im[3]);
    ((float4*)di)[1] = make_float4(im[4],im[5],im[6],im[7]);
  }
  #pragma unroll
  for (int vg=0; vg<8; ++vg){ re[vg] = pp1[vg]-qq1[vg]; im[vg] = rr1[vg]+ss1[vg]; }
  {
    float* dr = Utr + bcol1*PADD + rbase;
    float* di = Uti + bcol1*PADD + rbase;
    ((float4*)dr)[0] = make_float4(re[0],re[1],re[2],re[3]);
    ((float4*)dr)[1] = make_float4(re[4],re[5],re[6],re[7]);
    ((float4*)di)[0] = make_float4(im[0],im[1],im[2],im[3]);
    ((float4*)di)[1] = make_float4(im[4],im[5],im[6],im[7]);
  }
}

__global__ void __launch_bounds__(32)
k_cgemm_s2(const float* __restrict__ Er, const float* __restrict__ Ei,
           const float* __restrict__ Utr, const float* __restrict__ Uti,
           float* __restrict__ Or_, float* __restrict__ Oi){
  const int lane = threadIdx.x;
  const int rm = blockIdx.y * 16;
  const int cn = blockIdx.x * 32;
  const int half = lane >> 4;
  const int l15  = lane & 15;
  const size_t arow  = rm + l15;
  const size_t bcol0 = cn + l15;
  const size_t bcol1 = cn + 16 + l15;
  v8f z8 = {0.f,0.f,0.f,0.f,0.f,0.f,0.f,0.f};
  v8f pp0=z8, qq0=z8, rr0=z8, ss0=z8;
  v8f pp1=z8, qq1=z8, rr1=z8, ss1=z8;
  const float* ArRow = Er + arow*PADD;
  const float* AiRow = Ei + arow*PADD;
  const float* Br0   = Utr + bcol0*PADD;    // B[k][n] = Ut[n][k]
  const float* Bi0   = Uti + bcol0*PADD;
  const float* Br1   = Utr + bcol1*PADD;
  const float* Bi1   = Uti + bcol1*PADD;
  for (int k0 = 0; k0 < PADD; k0 += 4){
    const int ka = k0 + half*2;
    __builtin_prefetch(Br0 + ka + 64, 0, 1);
    v2f ar  = { ArRow[ka], ArRow[ka+1] };
    v2f ai  = { AiRow[ka], AiRow[ka+1] };
    v2f br0 = { Br0[ka],   Br0[ka+1]   };
    v2f bi0 = { Bi0[ka],   Bi0[ka+1]   };
    v2f br1 = { Br1[ka],   Br1[ka+1]   };
    v2f bi1 = { Bi1[ka],   Bi1[ka+1]   };
    pp0 = __builtin_amdgcn_wmma_f32_16x16x4_f32(false, ar, false, br0, (short)0, pp0, false, false);
    qq0 = __builtin_amdgcn_wmma_f32_16x16x4_f32(false, ai, false, bi0, (short)0, qq0, false, false);
    rr0 = __builtin_amdgcn_wmma_f32_16x16x4_f32(false, ar, false, bi0, (short)0, rr0, false, false);
    ss0 = __builtin_amdgcn_wmma_f32_16x16x4_f32(false, ai, false, br0, (short)0, ss0, false, false);
    pp1 = __builtin_amdgcn_wmma_f32_16x16x4_f32(false, ar, false, br1, (short)0, pp1, false, false);
    qq1 = __builtin_amdgcn_wmma_f32_16x16x4_f32(false, ai, false, bi1, (short)0, qq1, false, false);
    rr1 = __builtin_amdgcn_wmma_f32_16x16x4_f32(false, ar, false, bi1, (short)0, rr1, false, false);
    ss1 = __builtin_amdgcn_wmma_f32_16x16x4_f32(false, ai, false, br1, (short)0, ss1, false, false);
  }
  #pragma unroll
  for (int vg = 0; vg < 8; ++vg){
    const size_t row = rm + vg + half*8;
    Or_[row*PADD + bcol0] = pp0[vg] - qq0[vg];
    Oi [row*PADD + bcol0] = rr0[vg] + ss0[vg];
    Or_[row*PADD + bcol1] = pp1[vg] - qq1[vg];
    Oi [row*PADD + bcol1] = rr1[vg] + ss1[vg];
  }
}

// ================= |O| / (M*N) -> outGNN =================
__global__ void __launch_bounds__(256)
k_abs(const float* __restrict__ Or_, const float* __restrict__ Oi, float* __restrict__ out){
  int idx = blockIdx.x*256 + threadIdx.x;
  if (idx >= MN) return;
  int u = idx / NDIM, v = idx - u*NDIM;
  float a = Or_[(size_t)u*PADD + v], b = Oi[(size_t)u*PADD + v];
  out[idx] = sqrtf(fmaf(a, a, b*b)) * (1.0f / 591361.0f);
}

// ================= guided filter: vertical box height 2*KS+1, zero pad =================
__global__ void __launch_bounds__(256)
k_guided(const float* __restrict__ Xd, const float* __restrict__ gnn, float* __restrict__ outLR){
  int idx = blockIdx.x*256 + threadIdx.x;
  if (idx >= MN) return;
  int u = idx / NDIM, v = idx - u*NDIM;
  int lo = u - KS; if (lo < 0) lo = 0;
  int hi = u + KS; if (hi > MDIM-1) hi = MDIM-1;
  float s = 0.f;
  for (int w = lo; w <= hi; ++w){
    float g = gnn[(size_t)w*NDIM + v];
    s += fmaxf(g, 0.f) - Xd[(size_t)w*NDIM + v];
  }
  outLR[idx] = Xd[idx] + s / (float)(hi - lo + 1);
}

// ================= launch =================
extern "C" void kernel_launch(void* const* d_in, const int* in_sizes, int n_in,
                              void* d_out, int out_size, void* d_ws, size_t ws_size,
                              hipStream_t stream){
  (void)in_sizes; (void)n_in; (void)out_size; (void)ws_size;
  // inputs flattened in setup_inputs() dict order (params dict flattened in insertion order)
  const float* Xd   = (const float*)d_in[0];
  const c32*   Xf   = (const c32*)  d_in[1];
  const c32 *pf1W=(const c32*)d_in[2],  *pf1b=(const c32*)d_in[3];
  const c32 *pf2W=(const c32*)d_in[4],  *pf2b=(const c32*)d_in[5];
  const c32 *pskW=(const c32*)d_in[6],  *pskb=(const c32*)d_in[7];
  const c32 *e1W =(const c32*)d_in[8],  *e1b =(const c32*)d_in[9];
  const c32 *e2W =(const c32*)d_in[10], *e2b =(const c32*)d_in[11];
  const c32 *l1W =(const c32*)d_in[12], *l1b =(const c32*)d_in[13];
  const c32 *l2W =(const c32*)d_in[14], *l2b =(const c32*)d_in[15];
  const c32 *m1W =(const c32*)d_in[16], *m1b =(const c32*)d_in[17];
  const c32 *m2W =(const c32*)d_in[18], *m2b =(const c32*)d_in[19];
  const c32 *m3W =(const c32*)d_in[20], *m3b =(const c32*)d_in[21];
  const float *b1W=(const float*)d_in[22], *b1b=(const float*)d_in[23];
  const float *b2W=(const float*)d_in[24], *b2b=(const float*)d_in[25];
  const float *b3W=(const float*)d_in[26], *b3b=(const float*)d_in[27];
  const float *wco=(const float*)d_in[28];
  const int *NI   =(const int*)d_in[29];
  const int *hmask=(const int*)d_in[30];
  const int *hind =(const int*)d_in[31];

  // workspace layout (floats)
  float* ws = (float*)d_ws;
  constexpr size_t OFF_XFP = 0;                         // K*16 complex
  constexpr size_t OFF_Z   = OFF_XFP + (size_t)KC*32;   // K complex
  constexpr size_t OFF_S   = OFF_Z   + (size_t)KC*2;    // 1 float (+pad)
  constexpr size_t OFF_PRT = OFF_S   + 16;              // 256 partials
  constexpr size_t OFF_ER  = OFF_PRT + 256;
  constexpr size_t OFF_EI  = OFF_ER  + PP;
  constexpr size_t OFF_ZR  = OFF_EI  + PP;
  constexpr size_t OFF_ZI  = OFF_ZR  + PP;
  constexpr size_t OFF_UTR = OFF_ZI  + PP;
  constexpr size_t OFF_UTI = OFF_UTR + PP;
  constexpr size_t OFF_OR  = OFF_UTI + PP;
  constexpr size_t OFF_OI  = OFF_OR  + PP;
  c32*   Xfp = (c32*)(ws + OFF_XFP);
  c32*   z   = (c32*)(ws + OFF_Z);
  float* S   = ws + OFF_S;
  float* prt = ws + OFF_PRT;
  float* Er  = ws + OFF_ER;   float* Ei  = ws + OFF_EI;
  float* Zr  = ws + OFF_ZR;   float* Zi  = ws + OFF_ZI;
  float* Utr = ws + OFF_UTR;  float* Uti = ws + OFF_UTI;
  float* Or_ = ws + OFF_OR;   float* Oi  = ws + OFF_OI;

  float* outGNN = (float*)d_out;
  float* outLR  = outGNN + MN;

  const int gK  = (KC + 255) / 256;   // 1155
  const int gPP = (PP + 255) / 256;   // 2500
  const int gMN = (MN + 255) / 256;   // 2311

  k_pointwise<<<gK, 256, 0, stream>>>(Xf, pf1W, pf1b, pf2W, pf2b, pskW, pskb, Xfp);
  k_main<<<gK, 256, 0, stream>>>(Xfp, z, NI, wco, hmask, hind,
                                 e1W, e1b, e2W, e2b, l1W, l1b, l2W, l2b,
                                 m1W, m1b, m2W, m2b, m3W, m3b);
  k_sum_stage1<<<256, 256, 0, stream>>>(Xd, prt);
  k_sum_stage2<<<1, 256, 0, stream>>>(prt, S);
  k_dft_e<<<gPP, 256, 0, stream>>>(Er, Ei);
  k_build_zf<<<gPP, 256, 0, stream>>>(z, S, b1W, b1b, b2W, b2b, b3W, b3b, Zr, Zi);
  dim3 gg(PADD/32, PADD/16);                               // 25 x 50 waves
  k_cgemm_s1<<<gg, 32, 0, stream>>>(Zr, Zi, Er, Ei, Utr, Uti);   // Ut = (Z*E)^T
  k_cgemm_s2<<<gg, 32, 0, stream>>>(Er, Ei, Utr, Uti, Or_, Oi);  // O  = E*U
  k_abs<<<gMN, 256, 0, stream>>>(Or_, Oi, outGNN);
  k_guided<<<gMN, 256, 0, stream>>>(Xd, outGNN, outLR);
}